// GraphAttentionLayer_80659485819621
// MI455X (gfx1250) — compile-verified
//
#include <hip/hip_runtime.h>
#include <hip/hip_bf16.h>

typedef float v2f __attribute__((ext_vector_type(2)));
typedef float v8f __attribute__((ext_vector_type(8)));

#define GAT_B 4
#define GAT_N 2048
#define GAT_FIN 128
#define GAT_H 4
#define GAT_DK 32
#define LRELU_ALPHA 0.2f
#define NEG_BIG (-1.0e9f)

__device__ __forceinline__ unsigned ballot32(bool c) {
#if __has_builtin(__builtin_amdgcn_ballot_w32)
  return __builtin_amdgcn_ballot_w32(c);
#else
  unsigned v = c ? (1u << (threadIdx.x & 31)) : 0u;
#pragma unroll
  for (int o = 16; o; o >>= 1) v |= __shfl_xor(v, o);
  return v;
#endif
}

// ---------------------------------------------------------------------------
// Kernel A: h[b,h,n,d] = sum_i x[b,n,i] * W[h,i,d];  el/er = <h, a_l/a_r>
// One wave per node n; lane == d (DK == 32 == wave32 width).
// grid = B*H*(N/8), block = 256 (8 waves -> 8 nodes)
// ---------------------------------------------------------------------------
__global__ void __launch_bounds__(256)
gat_proj_kernel(const float* __restrict__ x, const float* __restrict__ W,
                const float* __restrict__ a, float* __restrict__ hbuf,
                float* __restrict__ elg, float* __restrict__ erg) {
  const int NT = GAT_N / 8;
  int blk = blockIdx.x;
  int ntile = blk % NT;
  int bh = blk / NT;
  int hh = bh % GAT_H;
  int b = bh / GAT_H;
  int tid = threadIdx.x;
  int w = tid >> 5;       // wave -> local node
  int lane = tid & 31;    // lane -> d
  int n = ntile * 8 + w;

  __shared__ float sW[GAT_FIN * GAT_DK];  // 16 KB: W[hh]
  __shared__ float sX[8 * GAT_FIN];       // 4 KB: 8 x-rows

  const float* Wh = W + (size_t)hh * GAT_FIN * GAT_DK;
  for (int t = tid; t < GAT_FIN * GAT_DK / 4; t += 256)
    ((float4*)sW)[t] = ((const float4*)Wh)[t];
  const float* xb = x + ((size_t)b * GAT_N + (size_t)ntile * 8) * GAT_FIN;
  for (int t = tid; t < 8 * GAT_FIN / 4; t += 256)
    ((float4*)sX)[t] = ((const float4*)xb)[t];
  __syncthreads();

  float acc = 0.0f;
#pragma unroll 8
  for (int i = 0; i < GAT_FIN; ++i)
    acc = fmaf(sX[w * GAT_FIN + i], sW[i * GAT_DK + lane], acc);

  hbuf[((size_t)bh * GAT_N + n) * GAT_DK + lane] = acc;

  float vl = acc * a[hh * 2 * GAT_DK + lane];
  float vr = acc * a[hh * 2 * GAT_DK + GAT_DK + lane];
#pragma unroll
  for (int off = 16; off; off >>= 1) {
    vl += __shfl_xor(vl, off);
    vr += __shfl_xor(vr, off);
  }
  if (lane == 0) {
    elg[(size_t)bh * GAT_N + n] = vl;
    erg[(size_t)bh * GAT_N + n] = vr;
  }
}

// ---------------------------------------------------------------------------
// Kernel B: masked-softmax attention + alpha@h via V_WMMA_F32_16X16X4_F32.
// block = 128 threads (4 waves); each wave owns a 16-row tile.
// grid = B*H*(N/64).
// ---------------------------------------------------------------------------
#define KB_WAVES 4
#define KB_ROWS (KB_WAVES * 16)  // 64 rows per block
#define SH_STRIDE 48             // padded h-tile row stride (bank-conflict free)
#define MASK_STRIDE 65           // padded mask row stride (words)

__global__ void __launch_bounds__(128)
gat_attn_kernel(const int* __restrict__ adj, const float* __restrict__ hbuf,
                const float* __restrict__ elg, const float* __restrict__ erg,
                float* __restrict__ out) {
  const int NRB = GAT_N / KB_ROWS;  // 32 row-blocks per (b,h)
  int blk = blockIdx.x;
  int rowblk = blk % NRB;
  int bh = blk / NRB;
  int hh = bh % GAT_H;
  int b = bh / GAT_H;
  int tid = threadIdx.x;
  int w = tid >> 5;
  int lane = tid & 31;
  int r0 = rowblk * KB_ROWS + w * 16;  // wave's first global row

  __shared__ float s_er[GAT_N];                       // 8 KB
  __shared__ float s_el[KB_ROWS];
  __shared__ float s_rowm[KB_ROWS];
  __shared__ float s_rowinv[KB_ROWS];
  __shared__ unsigned s_mask[KB_ROWS * MASK_STRIDE];  // ~16.6 KB adjacency bits
  __shared__ float s_h[32 * SH_STRIDE];               // 6 KB h chunk

  const float* erp = erg + (size_t)bh * GAT_N;
  for (int t = tid; t < GAT_N; t += KB_WAVES * 32) s_er[t] = erp[t];
  const float* elp = elg + (size_t)bh * GAT_N + (size_t)rowblk * KB_ROWS;
  if (tid < KB_ROWS) s_el[tid] = elp[tid];
  __syncthreads();

  // ------------------ Phase 1: online softmax stats, pack adj bits ---------
  float m[16], s[16], elr[16];
#pragma unroll
  for (int r = 0; r < 16; ++r) {
    m[r] = -3.0e38f;
    s[r] = 0.0f;
    elr[r] = s_el[w * 16 + r];
  }
  const int* adjrow = adj + (size_t)b * GAT_N * GAT_N + (size_t)r0 * GAT_N;
  for (int jc = 0; jc < GAT_N; jc += 32) {
    float erv = s_er[jc + lane];
#pragma unroll
    for (int r = 0; r < 16; ++r) {
      int ad = adjrow[(size_t)r * GAT_N + jc + lane];
      float e = elr[r] + erv;
      e = (e > 0.0f) ? e : LRELU_ALPHA * e;
      bool con = (ad != 0);
      e = con ? e : NEG_BIG;
      unsigned bal = ballot32(con);
      if (lane == 0) s_mask[(w * 16 + r) * MASK_STRIDE + (jc >> 5)] = bal;
      float mn = fmaxf(m[r], e);
      s[r] = s[r] * __expf(m[r] - mn) + __expf(e - mn);
      m[r] = mn;
    }
  }
  // cross-lane combine of (m, s) per row
#pragma unroll
  for (int r = 0; r < 16; ++r) {
    float mr = m[r], sr = s[r];
#pragma unroll
    for (int off = 16; off; off >>= 1) {
      float mo = __shfl_xor(mr, off);
      float so = __shfl_xor(sr, off);
      float mn = fmaxf(mr, mo);
      sr = sr * __expf(mr - mn) + so * __expf(mo - mn);
      mr = mn;
    }
    if (lane == 0) {
      s_rowm[w * 16 + r] = mr;
      s_rowinv[w * 16 + r] = 1.0f / sr;
    }
  }
  // same-wave LDS write->read is in-order; cross-wave use is only s_h (barriered)

  // ------------------ Phase 2: D += P(16xK) x H(Kx16) via f32 WMMA ---------
  // ISA 7.12.2 layouts (wave32):
  //  A 16x4 : lane l, vgpr v -> (M = l%16, K = v + 2*(l/16))
  //  B 4x16 : lane l, vgpr v -> (K = v + 2*(l/16), N = l%16)
  //  C/D    : lane l, vgpr v -> (M = v + 8*(l/16), N = l%16)
  v8f acc0 = {};  // d-columns 0..15
  v8f acc1 = {};  // d-columns 16..31
  int mm = lane & 15;
  int kk = (lane >> 4) << 1;
  float el_m = s_el[w * 16 + mm];
  float rowm_m = s_rowm[w * 16 + mm];

  for (int jc = 0; jc < GAT_N; jc += 32) {
    __syncthreads();  // previous chunk fully consumed
    {  // cooperative stage of h rows jc..jc+31 (32x32 f32)
      const float* hp = hbuf + ((size_t)bh * GAT_N + jc) * GAT_DK;
      int row = tid >> 2;
      int cg = (tid & 3) * 8;
      float4 p0 = *(const float4*)(hp + row * GAT_DK + cg);
      float4 p1 = *(const float4*)(hp + row * GAT_DK + cg + 4);
      *(float4*)(&s_h[row * SH_STRIDE + cg]) = p0;
      *(float4*)(&s_h[row * SH_STRIDE + cg + 4]) = p1;
    }
    __syncthreads();

    unsigned mw = s_mask[(w * 16 + mm) * MASK_STRIDE + (jc >> 5)];
#pragma unroll
    for (int ks = 0; ks < 8; ++ks) {
      int j0 = jc + ks * 4;
      int jA = j0 + kk;
      // A fragment: unnormalized probabilities p = exp(e - m_row)
      v2f A;
#pragma unroll
      for (int v = 0; v < 2; ++v) {
        int j = jA + v;
        float e = el_m + s_er[j];
        e = (e > 0.0f) ? e : LRELU_ALPHA * e;
        e = ((mw >> (j & 31)) & 1u) ? e : NEG_BIG;
        A[v] = __expf(e - rowm_m);
      }
      // B fragments from staged h tile
      int rb = jA - jc;
      v2f B0, B1;
      B0[0] = s_h[rb * SH_STRIDE + mm];
      B0[1] = s_h[(rb + 1) * SH_STRIDE + mm];
      B1[0] = s_h[rb * SH_STRIDE + 16 + mm];
      B1[1] = s_h[(rb + 1) * SH_STRIDE + 16 + mm];
      acc0 = __builtin_amdgcn_wmma_f32_16x16x4_f32(false, A, false, B0,
                                                   (short)0, acc0, false, false);
      acc1 = __builtin_amdgcn_wmma_f32_16x16x4_f32(false, A, false, B1,
                                                   (short)0, acc1, false, false);
    }
  }

  // ------------------ Epilogue: scale rows by 1/s and store ----------------
  int rowoff = 8 * (lane >> 4);
  size_t obase = ((size_t)b * GAT_N + r0) * (GAT_H * GAT_DK) + (size_t)hh * GAT_DK;
#pragma unroll
  for (int v = 0; v < 8; ++v) {
    int rr = v + rowoff;
    float inv = s_rowinv[w * 16 + rr];
    out[obase + (size_t)rr * (GAT_H * GAT_DK) + mm] = acc0[v] * inv;
    out[obase + (size_t)rr * (GAT_H * GAT_DK) + 16 + mm] = acc1[v] * inv;
  }
}

// ---------------------------------------------------------------------------
extern "C" void kernel_launch(void* const* d_in, const int* in_sizes, int n_in,
                              void* d_out, int out_size, void* d_ws, size_t ws_size,
                              hipStream_t stream) {
  (void)in_sizes; (void)n_in; (void)out_size; (void)ws_size;
  const float* x = (const float*)d_in[0];
  const int* adj = (const int*)d_in[1];
  const float* W = (const float*)d_in[2];
  const float* a = (const float*)d_in[3];
  float* out = (float*)d_out;

  // workspace: h (4 MB) | el (128 KB) | er (128 KB)
  float* hbuf = (float*)d_ws;
  float* elg = hbuf + (size_t)GAT_B * GAT_H * GAT_N * GAT_DK;
  float* erg = elg + (size_t)GAT_B * GAT_H * GAT_N;

  dim3 gridA(GAT_B * GAT_H * (GAT_N / 8));
  gat_proj_kernel<<<gridA, 256, 0, stream>>>(x, W, a, hbuf, elg, erg);

  dim3 gridB(GAT_B * GAT_H * (GAT_N / KB_ROWS));
  gat_attn_kernel<<<gridB, 128, 0, stream>>>(adj, hbuf, elg, erg, out);
}